// BasicBlock_15753940042155
// MI455X (gfx1250) — compile-verified
//
#include <hip/hip_runtime.h>
#include <hip/hip_bf16.h>

// ---------------------------------------------------------------------------
// AdderNet BasicBlock for MI455X (gfx1250, wave32).
//   conv = -sum |patch - w|  (pure VALU SAD, LDS-tiled, async global->LDS)
//   SE fc layers use V_WMMA_F32_16X16X4_F32 (exact fp32 matrix pipe).
// ---------------------------------------------------------------------------

#define NN 16
#define CC 128
#define HW 32
#define PLANE (HW * HW)        // 1024
#define CO_TILE 8
#define CI_CHUNK 8
#define TILE_W 34              // 32 + halo
#define TILE_PIX (TILE_W * TILE_W)  // 1156
#define BORDER_CELLS 132       // 2*34 + 2*32 halo cells per channel tile
#define EPS 1e-5f

typedef float v2f __attribute__((ext_vector_type(2)));
typedef float v8f __attribute__((ext_vector_type(8)));

#if __has_builtin(__builtin_amdgcn_global_load_async_to_lds_b32) && \
    __has_builtin(__builtin_amdgcn_s_wait_asynccnt)
#define HAVE_ASYNC_LDS 1
typedef __attribute__((address_space(1))) int g1int;
typedef __attribute__((address_space(3))) int l3int;
#else
#warning "gfx1250 async-to-LDS builtins not available on this toolchain; using sync staging fallback"
#endif

// ---------------------------------------------------------------------------
// Adder conv (SAD) + BatchNorm (+optional ReLU) + optional per-channel mean.
// Block: 256 threads. blockIdx.x = n, blockIdx.y = co tile (8 channels).
// Each thread owns 4 pixels x 8 channels = 32 f32 accumulators (high ILP for
// the v_sub_f32 / v_add_f32(|..|) stream that dominates this workload).
// The zero halo border is written once (invariant across ci chunks) and only
// interior cells are streamed per chunk via GLOBAL_LOAD_ASYNC_TO_LDS_B32.
// ---------------------------------------------------------------------------
__global__ __launch_bounds__(256) void adder_bn_kernel(
    const float* __restrict__ x,       // [N,C,32,32]
    const float* __restrict__ w,       // [C,C,3,3]
    const float* __restrict__ gamma,
    const float* __restrict__ beta,
    const float* __restrict__ mean,
    const float* __restrict__ var,
    float* __restrict__ out,           // [N,C,32,32]
    float* __restrict__ smean,         // [N,C] or nullptr
    int relu_flag)
{
    const int n   = blockIdx.x;
    const int co0 = blockIdx.y * CO_TILE;
    const int tid = threadIdx.x;

    __shared__ float lx[CI_CHUNK * TILE_PIX];            // 36992 B
    __shared__ float lw[CO_TILE * CI_CHUNK * 9];         //  2304 B
    __shared__ float lred[CO_TILE][8];

    // 4 pixels per thread: pix = tid + p*256 covers the full 32x32 plane.
    int py[4], px[4];
#pragma unroll
    for (int p = 0; p < 4; ++p) {
        int pix = tid + p * 256;
        py[p] = pix >> 5;
        px[p] = pix & 31;
    }

    float acc[CO_TILE][4];
#pragma unroll
    for (int co = 0; co < CO_TILE; ++co)
#pragma unroll
        for (int p = 0; p < 4; ++p) acc[co][p] = 0.f;

    // ---- zero the halo border once (padding contributes |0 - w|, matching
    //      the reference's zero-padded patches; identical for every chunk) ----
    for (int idx = tid; idx < CI_CHUNK * BORDER_CELLS; idx += 256) {
        int ci = idx / BORDER_CELLS;
        int b  = idx - ci * BORDER_CELLS;
        int ty, tx;
        if (b < 34)      { ty = 0;  tx = b; }
        else if (b < 68) { ty = 33; tx = b - 34; }
        else { int b2 = b - 68; ty = 1 + (b2 >> 1); tx = (b2 & 1) ? 33 : 0; }
        lx[ci * TILE_PIX + ty * TILE_W + tx] = 0.f;
    }

    for (int ci0 = 0; ci0 < CC; ci0 += CI_CHUNK) {
        __syncthreads();   // previous chunk's compute (and border fill) done

        // ---- stage interior x cells for this ci chunk ----
        const float* xp = x + ((size_t)(n * CC + ci0)) * PLANE;  // 8 planes
        for (int idx = tid; idx < CI_CHUNK * PLANE; idx += 256) {
            int ci  = idx >> 10;
            int pix = idx & 1023;
            int li  = ci * TILE_PIX + ((pix >> 5) + 1) * TILE_W + (pix & 31) + 1;
#ifdef HAVE_ASYNC_LDS
            __builtin_amdgcn_global_load_async_to_lds_b32(
                (g1int*)(xp + idx), (l3int*)&lx[li], 0, 0);
#else
            lx[li] = xp[idx];
#endif
        }
        // ---- stage weights for this (co-tile, ci-chunk) ----
        for (int idx = tid; idx < CO_TILE * CI_CHUNK * 9; idx += 256) {
            int co  = idx / (CI_CHUNK * 9);
            int rem = idx - co * (CI_CHUNK * 9);
            int ci  = rem / 9;
            int tap = rem - ci * 9;
            lw[idx] = w[(((size_t)(co0 + co) * CC) + (ci0 + ci)) * 9 + tap];
        }
#ifdef HAVE_ASYNC_LDS
        __builtin_amdgcn_s_wait_asynccnt(0);
#endif
        __syncthreads();

        // ---- SAD inner loops ----
        for (int ci = 0; ci < CI_CHUNK; ++ci) {
            const float* lxc = &lx[ci * TILE_PIX];
#pragma unroll
            for (int ky = 0; ky < 3; ++ky) {
#pragma unroll
                for (int kx = 0; kx < 3; ++kx) {
                    float xv[4];
#pragma unroll
                    for (int p = 0; p < 4; ++p)
                        xv[p] = lxc[(py[p] + ky) * TILE_W + (px[p] + kx)];
#pragma unroll
                    for (int co = 0; co < CO_TILE; ++co) {
                        float wv = lw[(co * CI_CHUNK + ci) * 9 + ky * 3 + kx];
#pragma unroll
                        for (int p = 0; p < 4; ++p)
                            acc[co][p] += fabsf(xv[p] - wv);
                    }
                }
            }
        }
    }

    // ---- BN (+ReLU) epilogue + optional channel-mean ----
    float part[CO_TILE];
#pragma unroll
    for (int co = 0; co < CO_TILE; ++co) {
        int c = co0 + co;
        float inv  = gamma[c] * rsqrtf(var[c] + EPS);
        float bias = beta[c] - mean[c] * inv;
        float ps = 0.f;
        float* op = out + ((size_t)(n * CC + c)) * PLANE;
#pragma unroll
        for (int p = 0; p < 4; ++p) {
            float y = (-acc[co][p]) * inv + bias;   // bn(-sad)
            if (relu_flag) y = fmaxf(y, 0.f);
            op[tid + p * 256] = y;
            ps += y;
        }
        part[co] = ps;
    }

    if (smean != nullptr) {
        int wid = tid >> 5, lane = tid & 31;
#pragma unroll
        for (int co = 0; co < CO_TILE; ++co) {
            float v = part[co];
#pragma unroll
            for (int off = 16; off > 0; off >>= 1)
                v += __shfl_xor(v, off);
            if (lane == 0) lred[co][wid] = v;
        }
        __syncthreads();
        if (tid < CO_TILE) {
            float t = 0.f;
#pragma unroll
            for (int wv = 0; wv < 8; ++wv) t += lred[tid][wv];
            smean[n * CC + co0 + tid] = t * (1.f / (float)PLANE);
        }
    }
}

// ---------------------------------------------------------------------------
// SE gate via V_WMMA_F32_16X16X4_F32.
// One block, 8 waves. fc1: [16,128]x[128,8] (wave 0, 32 accumulating WMMAs).
// fc2: [16,8]x[8,128] (wave w handles N-tile w, 2 WMMAs).
// Per-lane layouts per ISA 7.12.2 (16x4 A, 4x16 B, 16x16 C/D, wave32):
//   A: lane -> m = lane&15, k pair = (lane>>4)*2
//   B: lane -> n = lane&15, k pair = (lane>>4)*2
//   C/D: vgpr r -> row = r + (lane>>4)*8, col = lane&15
// B columns 8..15 of fc1 are zeroed with a hoisted mask multiply (no exec
// toggling inside the WMMA loop).
// ---------------------------------------------------------------------------
__global__ __launch_bounds__(256) void se_gate_kernel(
    const float* __restrict__ s,      // [16,128]
    const float* __restrict__ fc1w,   // [8,128]
    const float* __restrict__ fc1b,   // [8]
    const float* __restrict__ fc2w,   // [128,8]
    const float* __restrict__ fc2b,   // [128]
    float* __restrict__ gate)         // [16,128]
{
    __shared__ float t[16][8];
    const int tid  = threadIdx.x;
    const int wid  = tid >> 5;
    const int lane = tid & 31;
    const int mn   = lane & 15;
    const int half = lane >> 4;

    if (wid == 0) {
        const float  msk = (mn < 8) ? 1.f : 0.f;
        const float* srow = s + mn * CC;
        const float* f1r  = fc1w + (mn & 7) * CC;
        v8f c = {};
#pragma unroll 1
        for (int k = 0; k < CC; k += 4) {
            int k0 = k + half * 2;
            v2f a, b;
            a.x = srow[k0];
            a.y = srow[k0 + 1];
            b.x = f1r[k0]     * msk;
            b.y = f1r[k0 + 1] * msk;
            c = __builtin_amdgcn_wmma_f32_16x16x4_f32(
                    false, a, false, b, (short)0, c, false, false);
        }
#pragma unroll
        for (int r = 0; r < 8; ++r) {
            int row = r + half * 8;
            if (mn < 8) t[row][mn] = fmaxf(c[r] + fc1b[mn], 0.f);
        }
    }
    __syncthreads();

    const int nglob = wid * 16 + mn;   // output channel for this lane's B col
    v8f c2 = {};
#pragma unroll
    for (int kb = 0; kb < 8; kb += 4) {
        int k0 = kb + half * 2;
        v2f a, b;
        a.x = t[mn][k0];
        a.y = t[mn][k0 + 1];
        b.x = fc2w[nglob * 8 + k0];
        b.y = fc2w[nglob * 8 + k0 + 1];
        c2 = __builtin_amdgcn_wmma_f32_16x16x4_f32(
                 false, a, false, b, (short)0, c2, false, false);
    }
    float bv = fc2b[nglob];
#pragma unroll
    for (int r = 0; r < 8; ++r) {
        int row = r + half * 8;       // batch index
        float g = 1.f / (1.f + expf(-(c2[r] + bv)));
        gate[row * CC + nglob] = g;
    }
}

// ---------------------------------------------------------------------------
// out = relu(y2 * gate[n,c] + x), vectorized float4 (2048 blocks x 256).
// ---------------------------------------------------------------------------
__global__ __launch_bounds__(256) void se_residual_kernel(
    const float4* __restrict__ y2,
    const float* __restrict__ gate,
    const float4* __restrict__ x,
    float4* __restrict__ out)
{
    int i  = blockIdx.x * 256 + threadIdx.x;   // float4 index; 256 per plane
    int nc = i >> 8;                           // (n*C + c)
    float g = gate[nc];
    float4 v  = y2[i];
    float4 xv = x[i];
    float4 r;
    r.x = fmaxf(v.x * g + xv.x, 0.f);
    r.y = fmaxf(v.y * g + xv.y, 0.f);
    r.z = fmaxf(v.z * g + xv.z, 0.f);
    r.w = fmaxf(v.w * g + xv.w, 0.f);
    out[i] = r;
}

// ---------------------------------------------------------------------------
extern "C" void kernel_launch(void* const* d_in, const int* in_sizes, int n_in,
                              void* d_out, int out_size, void* d_ws, size_t ws_size,
                              hipStream_t stream) {
    const float* x     = (const float*)d_in[0];
    const float* w1    = (const float*)d_in[1];
    const float* bn1g  = (const float*)d_in[2];
    const float* bn1b  = (const float*)d_in[3];
    const float* bn1m  = (const float*)d_in[4];
    const float* bn1v  = (const float*)d_in[5];
    const float* w2    = (const float*)d_in[6];
    const float* bn2g  = (const float*)d_in[7];
    const float* bn2b  = (const float*)d_in[8];
    const float* bn2m  = (const float*)d_in[9];
    const float* bn2v  = (const float*)d_in[10];
    const float* fc1w  = (const float*)d_in[11];
    const float* fc1b  = (const float*)d_in[12];
    const float* fc2w  = (const float*)d_in[13];
    const float* fc2b  = (const float*)d_in[14];
    float* out = (float*)d_out;

    const size_t elems = (size_t)NN * CC * PLANE;  // 2,097,152
    float* y1   = (float*)d_ws;
    float* y2   = y1 + elems;
    float* s    = y2 + elems;          // [16,128]
    float* gate = s + NN * CC;         // [16,128]

    dim3 cgrid(NN, CC / CO_TILE);      // (16, 16)

    adder_bn_kernel<<<cgrid, 256, 0, stream>>>(
        x, w1, bn1g, bn1b, bn1m, bn1v, y1, nullptr, 1);

    adder_bn_kernel<<<cgrid, 256, 0, stream>>>(
        y1, w2, bn2g, bn2b, bn2m, bn2v, y2, s, 0);

    se_gate_kernel<<<1, 256, 0, stream>>>(s, fc1w, fc1b, fc2w, fc2b, gate);

    se_residual_kernel<<<(int)(elems / 4 / 256), 256, 0, stream>>>(
        (const float4*)y2, gate, (const float4*)x, (float4*)out);
}